// FastIPGLayer_67164698575300
// MI455X (gfx1250) — compile-verified
//
#include <hip/hip_runtime.h>
#include <math.h>

static constexpr int BB = 8;
static constexpr int CC = 64;
static constexpr int HH = 192;
static constexpr int WW = 192;
static constexpr int PP = HH * WW;     // 36864
static constexpr int HD = 96;
static constexpr int WD = 96;
static constexpr int NG = 32;

typedef float v2f __attribute__((ext_vector_type(2)));
typedef float v8f __attribute__((ext_vector_type(8)));

// ---------------------------------------------------------------- init minmax
__global__ void k_init_minmax(unsigned int* mm) {
    int t = threadIdx.x;
    if (t < 8)       mm[t] = 0x7f800000u;  // +inf
    else if (t < 16) mm[t] = 0u;           // 0.0f (df >= 0)
}

// ---------------------------------------------------------------- 2x2 avg pool
__global__ void k_down(const float* __restrict__ x, float* __restrict__ down) {
    int idx = blockIdx.x * blockDim.x + threadIdx.x;
    if (idx >= BB * CC * HD * WD) return;
    int n  = idx % WD;
    int m  = (idx / WD) % HD;
    int bc = idx / (WD * HD);
    const float* p = x + (size_t)bc * PP + (2 * m) * WW + 2 * n;
    down[idx] = 0.25f * (p[0] + p[1] + p[WW] + p[WW + 1]);
}

// ---------------------------------------------------------------- df + min/max
__global__ void k_df(const float* __restrict__ x, const float* __restrict__ down,
                     float* __restrict__ df, unsigned int* __restrict__ mm) {
    __shared__ float smin[256], smax[256];
    int b   = blockIdx.y;
    int pix = blockIdx.x * 256 + threadIdx.x;
    int i = pix / WW, j = pix % WW;

    // bilinear 2x upsample taps (half-pixel centers, edge clamp == renormalize)
    int m = i >> 1, y0, y1; float wy0, wy1;
    if ((i & 1) == 0) { y0 = m - 1; if (y0 < 0) y0 = 0; y1 = m; wy0 = 0.25f; wy1 = 0.75f; }
    else              { y0 = m; y1 = m + 1; if (y1 > HD - 1) y1 = HD - 1; wy0 = 0.75f; wy1 = 0.25f; }
    int n = j >> 1, x0, x1; float wx0, wx1;
    if ((j & 1) == 0) { x0 = n - 1; if (x0 < 0) x0 = 0; x1 = n; wx0 = 0.25f; wx1 = 0.75f; }
    else              { x0 = n; x1 = n + 1; if (x1 > WD - 1) x1 = WD - 1; wx0 = 0.75f; wx1 = 0.25f; }

    const float* xb = x    + (size_t)b * CC * PP;
    const float* db = down + (size_t)b * CC * HD * WD;
    float acc = 0.f;
    for (int c = 0; c < CC; ++c) {
        float ctr = xb[(size_t)c * PP + pix];
        const float* dc = db + (size_t)c * HD * WD;
        float up = wy0 * (wx0 * dc[y0 * WD + x0] + wx1 * dc[y0 * WD + x1])
                 + wy1 * (wx0 * dc[y1 * WD + x0] + wx1 * dc[y1 * WD + x1]);
        acc += fabsf(ctr - up);
    }
    df[(size_t)b * PP + pix] = acc;

    smin[threadIdx.x] = acc; smax[threadIdx.x] = acc;
    __syncthreads();
    for (int s = 128; s > 0; s >>= 1) {
        if (threadIdx.x < s) {
            smin[threadIdx.x] = fminf(smin[threadIdx.x], smin[threadIdx.x + s]);
            smax[threadIdx.x] = fmaxf(smax[threadIdx.x], smax[threadIdx.x + s]);
        }
        __syncthreads();
    }
    if (threadIdx.x == 0) {
        atomicMin(&mm[b],     __float_as_uint(smin[0]));
        atomicMax(&mm[8 + b], __float_as_uint(smax[0]));
    }
}

// ---------------------------------------------------------------- GN statistics
__global__ void k_gnstat(const float* __restrict__ x, float* __restrict__ gst) {
    __shared__ float s1[256], s2[256];
    int bg = blockIdx.x;                 // b*32 + g
    int b = bg >> 5, g = bg & 31;
    const float* p = x + ((size_t)b * CC + g * 2) * PP;  // 2 contiguous channels
    float a = 0.f, q = 0.f;
    for (int i = threadIdx.x; i < 2 * PP; i += 256) {
        float v = p[i]; a += v; q += v * v;
    }
    s1[threadIdx.x] = a; s2[threadIdx.x] = q;
    __syncthreads();
    for (int s = 128; s > 0; s >>= 1) {
        if (threadIdx.x < s) {
            s1[threadIdx.x] += s1[threadIdx.x + s];
            s2[threadIdx.x] += s2[threadIdx.x + s];
        }
        __syncthreads();
    }
    if (threadIdx.x == 0) {
        const float invN = 1.0f / (2.0f * PP);
        float mean = s1[0] * invN;
        float var  = s2[0] * invN - mean * mean;
        gst[bg * 2]     = mean;
        gst[bg * 2 + 1] = rsqrtf(var + 1e-5f);
    }
}

// ------------------------------------------------------- neighborhood attention
#define CSW(i, j)                                                              \
    {                                                                          \
        bool sw = (sv[i] < sv[j]) || (sv[i] == sv[j] && id[i] > id[j]);        \
        if (sw) {                                                              \
            float ts = sv[i]; sv[i] = sv[j]; sv[j] = ts;                       \
            int   ti = id[i]; id[i] = id[j]; id[j] = ti;                       \
        }                                                                      \
    }

__global__ void k_attn(const float* __restrict__ x, const float* __restrict__ df,
                       const unsigned int* __restrict__ mm, const float* __restrict__ gst,
                       const float* __restrict__ gamma, const float* __restrict__ beta,
                       float* __restrict__ out) {
    __shared__ float ch[16 * 324];   // 16-channel chunk of 18x18 halo tile
    __shared__ float ssq[324];       // per-local-pixel sum of squares (all 64 ch)
    const int tid = threadIdx.x;
    const int b   = blockIdx.z;
    const int ty0 = blockIdx.y * 16, tx0 = blockIdx.x * 16;
    const int iy  = tid >> 4, ix = tid & 15;
    const int gy  = ty0 + iy, gx = tx0 + ix;
    const int lc  = (iy + 1) * 18 + (ix + 1);

    int lnb[9];
#pragma unroll
    for (int di = 0; di < 3; ++di)
#pragma unroll
        for (int dj = 0; dj < 3; ++dj)
            lnb[di * 3 + dj] = (iy + di) * 18 + (ix + dj);

    const float* xb = x + (size_t)b * CC * PP;

    float dot[9];
#pragma unroll
    for (int j = 0; j < 9; ++j) dot[j] = 0.f;
    for (int l = tid; l < 324; l += 256) ssq[l] = 0.f;

    // ---- phase A: dots + norms over 4 channel chunks
    for (int cb = 0; cb < 4; ++cb) {
        __syncthreads();
        for (int idx = tid; idx < 16 * 324; idx += 256) {
            int l = idx % 324, c = idx / 324;
            int ly = l / 18, lx = l % 18;
            int ry = ty0 + ly - 1; if (ry < 0) ry = -ry; if (ry >= HH) ry = 2 * HH - 2 - ry;
            int rx = tx0 + lx - 1; if (rx < 0) rx = -rx; if (rx >= WW) rx = 2 * WW - 2 - rx;
            ch[idx] = xb[(size_t)(cb * 16 + c) * PP + ry * WW + rx];
        }
        __syncthreads();
        for (int l = tid; l < 324; l += 256) {
            float s = 0.f;
#pragma unroll
            for (int c = 0; c < 16; ++c) { float v = ch[c * 324 + l]; s += v * v; }
            ssq[l] += s;
        }
#pragma unroll
        for (int c = 0; c < 16; ++c) {
            float ctr = ch[c * 324 + lc];
#pragma unroll
            for (int j = 0; j < 9; ++j) dot[j] += ctr * ch[c * 324 + lnb[j]];
        }
    }
    __syncthreads();

    float invc = 1.f / fmaxf(sqrtf(ssq[lc]), 1e-12f);
    float sv[9]; int id[9];
#pragma unroll
    for (int j = 0; j < 9; ++j) {
        float invn = 1.f / fmaxf(sqrtf(ssq[lnb[j]]), 1e-12f);
        sv[j] = dot[j] * invc * invn;
        id[j] = j;
    }
    // 9-element sorting network, descending, lower-index-first on ties (top_k)
    CSW(0,3) CSW(1,7) CSW(2,5) CSW(4,8)
    CSW(0,7) CSW(2,4) CSW(3,8) CSW(5,6)
    CSW(0,2) CSW(1,3) CSW(4,5) CSW(7,8)
    CSW(1,4) CSW(3,6) CSW(5,7)
    CSW(0,1) CSW(2,4) CSW(3,5) CSW(6,8)
    CSW(2,3) CSW(4,5) CSW(6,7)
    CSW(1,2) CSW(3,4) CSW(5,6)

    // variable k from df
    const int pix = gy * WW + gx;
    float dfv  = df[(size_t)b * PP + pix];
    float dmin = __uint_as_float(mm[b]);
    float dmax = __uint_as_float(mm[8 + b]);
    float dn = (dfv - dmin) / (dmax - dmin + 1e-8f);
    float dp = dn * dn; dp = dp * dp;
    int k = 1;
    if (dp > 0.9f) {
        float above = rintf((dp - 0.9f) / (1.0f - 0.9f + 1e-8f) * 15.0f);
        if (above < 0.f) above = 0.f;
        k = 1 + (int)above;
    }
    if (k > 9) k = 9;

    float wgt[9]; int lsel[9];
    float esum = 0.f;
#pragma unroll
    for (int j = 0; j < 9; ++j) {
        float e = (j < k) ? __expf(sv[j]) : 0.f;
        wgt[j] = e; esum += e;
    }
    float einv = 1.f / (esum + 1e-8f);
#pragma unroll
    for (int j = 0; j < 9; ++j) { wgt[j] *= einv; lsel[j] = lnb[id[j]]; }

    // ---- phase B: gather + fused GroupNorm, write enhanced
    for (int cb = 0; cb < 4; ++cb) {
        __syncthreads();
        for (int idx = tid; idx < 16 * 324; idx += 256) {
            int l = idx % 324, c = idx / 324;
            int ly = l / 18, lx = l % 18;
            int ry = ty0 + ly - 1; if (ry < 0) ry = -ry; if (ry >= HH) ry = 2 * HH - 2 - ry;
            int rx = tx0 + lx - 1; if (rx < 0) rx = -rx; if (rx >= WW) rx = 2 * WW - 2 - rx;
            ch[idx] = xb[(size_t)(cb * 16 + c) * PP + ry * WW + rx];
        }
        __syncthreads();
#pragma unroll
        for (int c = 0; c < 16; ++c) {
            int cg = cb * 16 + c;
            float acc = 0.f;
#pragma unroll
            for (int j = 0; j < 9; ++j) acc += wgt[j] * ch[c * 324 + lsel[j]];
            float ctr  = ch[c * 324 + lc];
            int   gi   = (b * NG + (cg >> 1)) * 2;
            float mean = gst[gi], istd = gst[gi + 1];
            out[((size_t)(b * CC + cg)) * PP + pix] =
                acc + (ctr - mean) * istd * gamma[cg] + beta[cg];
        }
    }
}

// ---------------------------------------------------------------- FFN via WMMA
// out (in/out) holds `enhanced` [B,C,H,W]; computes out += ffn(out) in place.
// One 16-pixel M-tile per wave (acc = 64 VGPRs -> no spilling), with the
// 8 B-fragments batched into a live bb[] array so the scheduler emits a
// load clause + partial waits + back-to-back v_wmma_f32_16x16x4_f32.
// A layout (f32 16x4): v0 = {K=k0 | K=k0+2}, v1 = {K=k0+1 | K=k0+3}, M = lane&15.
// D layout: vgpr j -> (M=j | M=j+8), N = lane&15.
__global__ void __launch_bounds__(128, 1)
k_ffn(float* __restrict__ out, const float* __restrict__ w1,
      const float* __restrict__ b1, const float* __restrict__ w2,
      const float* __restrict__ b2) {
    __shared__ float hT[4][16 * 130];  // per-wave h tile, 130-pad vs bank conflicts
    __shared__ float oT[4][64 * 17];   // per-wave transposed out tile, 17-pad
    const int lane = threadIdx.x & 31;
    const int wv   = threadIdx.x >> 5;
    float* h = hT[wv];
    float* o = oT[wv];

    const int tile = blockIdx.x * 4 + wv;   // one 16-pixel tile per wave
    const int gp0  = tile * 16;
    const int b    = gp0 / PP;
    const int p0   = gp0 % PP;
    const size_t base = (size_t)b * CC * PP + p0;
    const int mrow  = lane & 15;
    const int khalf = (lane >> 4) << 1;     // 0 or 2
    const int hi8   = (lane >> 4) << 3;     // 0 or 8

    // warm GL2/WGP$ with the shared weight matrices -> global_prefetch_b8
    __builtin_prefetch(w1 + threadIdx.x * 64, 0, 3);
    __builtin_prefetch(w2 + threadIdx.x * 64, 0, 3);

    // ---- GEMM1: [16 x 64] x [64 x 128] -> h
    v8f acc[8];
#pragma unroll
    for (int n = 0; n < 8; ++n) acc[n] = (v8f){0.f, 0.f, 0.f, 0.f, 0.f, 0.f, 0.f, 0.f};
    for (int k0 = 0; k0 < 64; k0 += 4) {
        int kk = k0 + khalf;
        v2f a;
        a.x = out[base + (size_t)kk * PP + mrow];
        a.y = out[base + (size_t)(kk + 1) * PP + mrow];
        v2f bb[8];
#pragma unroll
        for (int n = 0; n < 8; ++n) {
            int d = n * 16 + mrow;
            bb[n].x = w1[d * 64 + kk];
            bb[n].y = w1[d * 64 + kk + 1];
        }
#pragma unroll
        for (int n = 0; n < 8; ++n) {
            acc[n] = __builtin_amdgcn_wmma_f32_16x16x4_f32(
                false, a, false, bb[n], (short)0, acc[n], false, false);
        }
    }
    // bias + ReLU -> LDS (h)
#pragma unroll
    for (int n = 0; n < 8; ++n) {
        int d = n * 16 + mrow;
        float bias = b1[d];
#pragma unroll
        for (int j = 0; j < 8; ++j)
            h[(j + hi8) * 130 + d] = fmaxf(acc[n][j] + bias, 0.0f);
    }
    __syncthreads();

    // ---- GEMM2: [16 x 128] x [128 x 64] -> ffn
    v8f acc2[4];
#pragma unroll
    for (int n = 0; n < 4; ++n) acc2[n] = (v8f){0.f, 0.f, 0.f, 0.f, 0.f, 0.f, 0.f, 0.f};
    for (int k0 = 0; k0 < 128; k0 += 4) {
        int kk = k0 + khalf;
        v2f a;
        a.x = h[mrow * 130 + kk];
        a.y = h[mrow * 130 + kk + 1];
        v2f bb[4];
#pragma unroll
        for (int n = 0; n < 4; ++n) {
            int c = n * 16 + mrow;
            bb[n].x = w2[c * 128 + kk];
            bb[n].y = w2[c * 128 + kk + 1];
        }
#pragma unroll
        for (int n = 0; n < 4; ++n) {
            acc2[n] = __builtin_amdgcn_wmma_f32_16x16x4_f32(
                false, a, false, bb[n], (short)0, acc2[n], false, false);
        }
    }
    // bias -> transposed LDS staging
#pragma unroll
    for (int n = 0; n < 4; ++n) {
        int c = n * 16 + mrow;
        float bias = b2[c];
#pragma unroll
        for (int j = 0; j < 8; ++j)
            o[c * 17 + (j + hi8)] = acc2[n][j] + bias;
    }
    __syncthreads();

    // coalesced in-place residual writeback: out = enhanced + ffn
    for (int idx = lane; idx < 16 * 64; idx += 32) {
        int m = idx & 15, c = idx >> 4;
        size_t g = base + (size_t)c * PP + m;
        out[g] += o[c * 17 + m];
    }
}

// ---------------------------------------------------------------------- launch
extern "C" void kernel_launch(void* const* d_in, const int* in_sizes, int n_in,
                              void* d_out, int out_size, void* d_ws, size_t ws_size,
                              hipStream_t stream) {
    const float* x     = (const float*)d_in[0];
    const float* gamma = (const float*)d_in[1];
    const float* beta  = (const float*)d_in[2];
    const float* w1    = (const float*)d_in[3];
    const float* b1    = (const float*)d_in[4];
    const float* w2    = (const float*)d_in[5];
    const float* b2    = (const float*)d_in[6];
    float* out = (float*)d_out;

    // workspace layout (~20 MB)
    float*        ws_down = (float*)d_ws;                            // B*C*96*96
    float*        ws_df   = ws_down + (size_t)BB * CC * HD * WD;     // B*P
    unsigned int* ws_mm   = (unsigned int*)(ws_df + (size_t)BB * PP);// 16
    float*        ws_gst  = (float*)(ws_mm + 16);                    // B*32*2

    k_init_minmax<<<1, 64, 0, stream>>>(ws_mm);
    k_down<<<(BB * CC * HD * WD + 255) / 256, 256, 0, stream>>>(x, ws_down);
    k_df<<<dim3(PP / 256, BB), 256, 0, stream>>>(x, ws_down, ws_df, ws_mm);
    k_gnstat<<<BB * NG, 256, 0, stream>>>(x, ws_gst);
    k_attn<<<dim3(WW / 16, HH / 16, BB), 256, 0, stream>>>(
        x, ws_df, ws_mm, ws_gst, gamma, beta, out);
    // one 16-pixel tile per wave, 4 waves (128 threads) per block
    k_ffn<<<(BB * PP / 16) / 4, 128, 0, stream>>>(out, w1, b1, w2, b2);
}